// FirstMessage_50122268344442
// MI455X (gfx1250) — compile-verified
//
#include <hip/hip_runtime.h>
#include <hip/hip_bf16.h>

#define N_NODES   50000
#define N_EDGES   400000
#define NODE_DIM  128
#define NUM_BASIS 20
#define MUL0      128
#define MUL1      64
#define MUL2      32
#define NUM_IRREPS 224      // 128+64+32
#define SPH_DIM   480       // 128 + 3*64 + 5*32
#define HIDDEN    352       // 128 + 224
#define LN_EPS    1e-5f

typedef __attribute__((ext_vector_type(2))) float v2f;
typedef __attribute__((ext_vector_type(4))) float v4f;
typedef __attribute__((ext_vector_type(8))) float v8f;

// ---------------------------------------------------------------------------
// f32 WMMA: D(16x16) = A(16x4) * B(4x16) + C   (V_WMMA_F32_16X16X4_F32)
// ---------------------------------------------------------------------------
__device__ __forceinline__ v8f wmma_f32(v2f a, v2f b, v8f c) {
  return __builtin_amdgcn_wmma_f32_16x16x4_f32(
      /*neg_a=*/false, a, /*neg_b=*/false, b,
      /*c_mod=*/(short)0, c, /*reuse_a=*/false, /*reuse_b=*/false);
}

// native fp32 global atomic add (no-return form -> STOREcnt, no CAS loop).
// No "memory" clobber: adds commute, value operand carries the data dep,
// and S_ENDPGM's implicit wait-idle covers completion.
__device__ __forceinline__ void atomic_add_f32(float* p, float v) {
  asm volatile("global_atomic_add_f32 %0, %1, off" :: "v"(p), "v"(v));
}

__device__ __forceinline__ float wave_sum32(float v) {
  #pragma unroll
  for (int m = 16; m >= 1; m >>= 1) v += __shfl_xor(v, m, 32);
  return v;
}

// ---------------------------------------------------------------------------
// Kernel 1: per-node pipeline.  16 nodes per block, 128 threads (4 waves).
//   layer_norm -> GEMM1(128x128) -> SiLU -> GEMM2(128x352)
//   writes new_scalar init (= scalar_in) and scalar_out to workspace.
// ---------------------------------------------------------------------------
#define SSTR 132   // LDS row stride (floats) to dodge bank conflicts

__global__ void __launch_bounds__(128)
node_kernel(const float* __restrict__ x_scalar,
            const float* __restrict__ W1, const float* __restrict__ b1,
            const float* __restrict__ W2, const float* __restrict__ b2,
            const float* __restrict__ ln_g, const float* __restrict__ ln_b,
            float* __restrict__ out_scalar,   // [N_NODES,128] init
            float* __restrict__ scalar_out)   // ws [N_NODES,352]
{
  __shared__ float sIn[16 * SSTR];
  __shared__ float sH [16 * SSTR];
  __shared__ float sMu[16], sRs[16];

  const int tid  = threadIdx.x;
  const int wave = tid >> 5;
  const int lane = tid & 31;
  const int half = lane >> 4;
  const int lr   = lane & 15;
  const int node0 = blockIdx.x * 16;

  // ---- load 16x128 tile ----
  for (int i = tid; i < 16 * NODE_DIM; i += 128) {
    int r = i >> 7, c = i & 127;
    sIn[r * SSTR + c] = x_scalar[(size_t)(node0 + r) * NODE_DIM + c];
  }
  __syncthreads();

  // ---- layer-norm stats (one thread per row) ----
  if (tid < 16) {
    float s = 0.f, ss = 0.f;
    #pragma unroll 4
    for (int c = 0; c < NODE_DIM; ++c) {
      float v = sIn[tid * SSTR + c];
      s += v; ss += v * v;
    }
    float mu  = s * (1.0f / NODE_DIM);
    float var = ss * (1.0f / NODE_DIM) - mu * mu;
    sMu[tid] = mu;
    sRs[tid] = rsqrtf(var + LN_EPS);
  }
  __syncthreads();

  // ---- normalize; write scalar_in to LDS and to output (init) ----
  for (int i = tid; i < 16 * NODE_DIM; i += 128) {
    int r = i >> 7, c = i & 127;
    float v = (sIn[r * SSTR + c] - sMu[r]) * sRs[r] * ln_g[c] + ln_b[c];
    sIn[r * SSTR + c] = v;
    out_scalar[(size_t)(node0 + r) * NODE_DIM + c] = v;
  }
  __syncthreads();

  // ---- GEMM1: h = silu(scalar_in @ W1 + b1), 8 N-tiles over 4 waves ----
  for (int nt = wave; nt < NODE_DIM / 16; nt += 4) {
    const int n = nt * 16 + lr;
    float bv = b1[n];
    v8f c = {bv, bv, bv, bv, bv, bv, bv, bv};
    #pragma unroll 8
    for (int kb = 0; kb < NODE_DIM; kb += 4) {
      v2f a = *(const v2f*)&sIn[lr * SSTR + kb + 2 * half];
      v2f b;
      b.x = W1[(kb + 2 * half)     * NODE_DIM + n];
      b.y = W1[(kb + 2 * half + 1) * NODE_DIM + n];
      c = wmma_f32(a, b, c);
    }
    #pragma unroll
    for (int i = 0; i < 8; ++i) {
      float h = c[i];
      // silu via fast rcp: one v_exp + one v_rcp instead of IEEE div sequence
      h = h * __builtin_amdgcn_rcpf(1.0f + __expf(-h));
      sH[(i + 8 * half) * SSTR + n] = h;
    }
  }
  __syncthreads();

  // ---- GEMM2: scalar_out = h @ W2 + b2, 22 N-tiles over 4 waves ----
  for (int nt = wave; nt < HIDDEN / 16; nt += 4) {
    const int n = nt * 16 + lr;
    float bv = b2[n];
    v8f c = {bv, bv, bv, bv, bv, bv, bv, bv};
    #pragma unroll 8
    for (int kb = 0; kb < NODE_DIM; kb += 4) {
      v2f a = *(const v2f*)&sH[lr * SSTR + kb + 2 * half];
      v2f b;
      b.x = W2[(kb + 2 * half)     * HIDDEN + n];
      b.y = W2[(kb + 2 * half + 1) * HIDDEN + n];
      c = wmma_f32(a, b, c);
    }
    #pragma unroll
    for (int i = 0; i < 8; ++i)
      scalar_out[(size_t)(node0 + i + 8 * half) * HIDDEN + n] = c[i];
  }
}

// ---------------------------------------------------------------------------
// Kernel 2: O(3) layer-norm of spherical features -> new_spherical init.
// One wave per node (wave32 shuffle reductions), 8 nodes per 256-thread block.
// ---------------------------------------------------------------------------
__global__ void __launch_bounds__(256)
sph_norm_kernel(const float* __restrict__ x_sph,
                const float* __restrict__ o3_w, const float* __restrict__ o3_b,
                float* __restrict__ out_sph)   // [N_NODES,480] init
{
  const int wave = threadIdx.x >> 5;
  const int lane = threadIdx.x & 31;
  const int node = blockIdx.x * 8 + wave;
  const float* xs = x_sph + (size_t)node * SPH_DIM;
  float* os = out_sph + (size_t)node * SPH_DIM;

  // scalar irreps: 128 values (rows are 16B aligned: 480*4 % 16 == 0)
  v4f sv4 = *(const v4f*)(xs + lane * 4);
  float s  = sv4.x + sv4.y + sv4.z + sv4.w;
  float ss = sv4.x * sv4.x + sv4.y * sv4.y + sv4.z * sv4.z + sv4.w * sv4.w;
  s  = wave_sum32(s);
  ss = wave_sum32(ss);
  float mu = s * (1.0f / MUL0);
  float rs = rsqrtf(ss * (1.0f / MUL0) - mu * mu + LN_EPS);
  #pragma unroll
  for (int i = lane; i < MUL0; i += 32)
    os[i] = (xs[i] - mu) * rs * o3_w[i] + o3_b[i];

  // l=1 irreps: 64 x 3 = 192 values
  float sq1 = 0.f;
  #pragma unroll
  for (int k = 0; k < 6; ++k) {
    float v = xs[MUL0 + lane * 6 + k];
    sq1 += v * v;
  }
  sq1 = wave_sum32(sq1);
  float rv = rsqrtf(sq1 * (1.0f / (3 * MUL1)) + LN_EPS);
  #pragma unroll
  for (int i = lane; i < 3 * MUL1; i += 32) {
    int m = i / 3;
    os[MUL0 + i] = xs[MUL0 + i] * rv * o3_w[MUL0 + m];
  }

  // l=2 irreps: 32 x 5 = 160 values
  float sq2 = 0.f;
  #pragma unroll
  for (int k = 0; k < 5; ++k) {
    float v = xs[MUL0 + 3 * MUL1 + lane * 5 + k];
    sq2 += v * v;
  }
  sq2 = wave_sum32(sq2);
  float rt = rsqrtf(sq2 * (1.0f / (5 * MUL2)) + LN_EPS);
  #pragma unroll
  for (int i = lane; i < 5 * MUL2; i += 32) {
    int m = i / 5;
    os[MUL0 + 3 * MUL1 + i] = xs[MUL0 + 3 * MUL1 + i] * rt * o3_w[MUL0 + MUL1 + m];
  }
}

// ---------------------------------------------------------------------------
// Kernel 3: per-edge. 16 edges per block, 128 threads (4 waves).
//   fw = rbf @ Wrbf  (WMMA, K=20 = 5 steps of 4)
//   filter_out = scalar_out[src] * fw   (gather, L2-resident)
//   scatter gate*sph + message_scalar with native fp32 atomics.
// ---------------------------------------------------------------------------
#define FSTR 353   // LDS stride for 352-wide tile

__global__ void __launch_bounds__(128)
edge_kernel(const float* __restrict__ rbf, const float* __restrict__ vec,
            const float* __restrict__ Wrbf,
            const int* __restrict__ edge_index,
            const float* __restrict__ scalar_out,   // ws [N_NODES,352]
            float* __restrict__ out_scalar,         // [N_NODES,128]
            float* __restrict__ out_sph)            // [N_NODES,480]
{
  __shared__ float s_rbf[16 * NUM_BASIS];
  __shared__ float s_filt[16 * FSTR];
  __shared__ float s_sh[16 * 8];
  __shared__ int   s_src[16], s_dst[16];

  const int tid  = threadIdx.x;
  const int wave = tid >> 5;
  const int lane = tid & 31;
  const int half = lane >> 4;
  const int lr   = lane & 15;
  const int e0   = blockIdx.x * 16;

  // ---- stage rbf tile, indices, spherical harmonics ----
  for (int i = tid; i < 16 * NUM_BASIS; i += 128)
    s_rbf[i] = rbf[(size_t)e0 * NUM_BASIS + i];
  if (tid < 16) {
    int e = e0 + tid;
    s_dst[tid] = edge_index[e];
    s_src[tid] = edge_index[N_EDGES + e];
    float vx = vec[e * 3 + 0], vy = vec[e * 3 + 1], vz = vec[e * 3 + 2];
    float rn = rsqrtf(vx * vx + vy * vy + vz * vz);
    float x = vx * rn, y = vy * rn, z = vz * rn;
    const float s3  = 1.7320508075688772f;
    const float s15 = 3.872983346207417f;
    const float s5  = 2.23606797749979f;
    s_sh[tid * 8 + 0] = s3 * x;
    s_sh[tid * 8 + 1] = s3 * y;
    s_sh[tid * 8 + 2] = s3 * z;
    s_sh[tid * 8 + 3] = s15 * x * z;
    s_sh[tid * 8 + 4] = s15 * x * y;
    s_sh[tid * 8 + 5] = s5 * (y * y - 0.5f * (x * x + z * z));
    s_sh[tid * 8 + 6] = s15 * y * z;
    s_sh[tid * 8 + 7] = 0.5f * s15 * (z * z - x * x);
  }
  __syncthreads();

  // ---- WMMA: fw(16x352) = rbf(16x20) @ Wrbf(20x352) ----
  for (int nt = wave; nt < HIDDEN / 16; nt += 4) {
    const int n = nt * 16 + lr;
    v8f c = {0.f, 0.f, 0.f, 0.f, 0.f, 0.f, 0.f, 0.f};
    #pragma unroll
    for (int kb = 0; kb < NUM_BASIS; kb += 4) {
      v2f a = *(const v2f*)&s_rbf[lr * NUM_BASIS + kb + 2 * half];
      v2f b;
      b.x = Wrbf[(kb + 2 * half)     * HIDDEN + n];
      b.y = Wrbf[(kb + 2 * half + 1) * HIDDEN + n];
      c = wmma_f32(a, b, c);
    }
    #pragma unroll
    for (int i = 0; i < 8; ++i)
      s_filt[(i + 8 * half) * FSTR + n] = c[i];
  }
  __syncthreads();

  // ---- gather scalar_out[src] and multiply in place (rows L2-resident) ----
  for (int e = 0; e < 16; ++e) {
    const float* row = scalar_out + (size_t)s_src[e] * HIDDEN;
    for (int d = tid; d < HIDDEN; d += 128)
      s_filt[e * FSTR + d] *= row[d];
  }
  __syncthreads();

  // ---- scatter: 608 atomic fp32 adds per edge ----
  for (int e = 0; e < 16; ++e) {
    const int dst = s_dst[e];
    float* oS = out_scalar + (size_t)dst * NODE_DIM;
    float* oV = out_sph    + (size_t)dst * SPH_DIM;
    const float* f  = &s_filt[e * FSTR];
    const float* sh = &s_sh[e * 8];

    // message_scalar (dims 224..351) and g0 (dims 0..127): tid == dim
    atomic_add_f32(oS + tid, f[NUM_IRREPS + tid]);
    atomic_add_f32(oV + tid, f[tid]);

    // l=1 messages: g1[m] * sh1[j], 192 values
    for (int r = tid; r < 3 * MUL1; r += 128) {
      int m = r / 3, j = r - 3 * m;
      atomic_add_f32(oV + MUL0 + r, f[MUL0 + m] * sh[j]);
    }
    // l=2 messages: g2[m] * sh2[j], 160 values
    for (int r = tid; r < 5 * MUL2; r += 128) {
      int m = r / 5, j = r - 5 * m;
      atomic_add_f32(oV + MUL0 + 3 * MUL1 + r, f[MUL0 + MUL1 + m] * sh[3 + j]);
    }
  }
}

// ---------------------------------------------------------------------------
extern "C" void kernel_launch(void* const* d_in, const int* in_sizes, int n_in,
                              void* d_out, int out_size, void* d_ws, size_t ws_size,
                              hipStream_t stream) {
  const float* x_scalar = (const float*)d_in[0];
  const float* x_sph    = (const float*)d_in[1];
  const float* rbf      = (const float*)d_in[2];
  const float* vec      = (const float*)d_in[3];
  const float* W1       = (const float*)d_in[4];
  const float* b1       = (const float*)d_in[5];
  const float* W2       = (const float*)d_in[6];
  const float* b2       = (const float*)d_in[7];
  const float* Wrbf     = (const float*)d_in[8];
  const float* ln_g     = (const float*)d_in[9];
  const float* ln_b     = (const float*)d_in[10];
  const float* o3_w     = (const float*)d_in[11];
  const float* o3_b     = (const float*)d_in[12];
  const int*   eidx     = (const int*)d_in[13];

  float* out_scalar = (float*)d_out;                               // [50000,128]
  float* out_sph    = out_scalar + (size_t)N_NODES * NODE_DIM;     // [50000,480]
  float* ws_so      = (float*)d_ws;                                // [50000,352]

  node_kernel<<<N_NODES / 16, 128, 0, stream>>>(
      x_scalar, W1, b1, W2, b2, ln_g, ln_b, out_scalar, ws_so);
  sph_norm_kernel<<<N_NODES / 8, 256, 0, stream>>>(
      x_sph, o3_w, o3_b, out_sph);
  edge_kernel<<<N_EDGES / 16, 128, 0, stream>>>(
      rbf, vec, Wrbf, eidx, ws_so, out_scalar, out_sph);
}